// LogMelSpecAugment_59442347376821
// MI455X (gfx1250) — compile-verified
//
#include <hip/hip_runtime.h>

// ---- problem constants (match reference) ----
#define B_DIM 64
#define F_DIM 128
#define T_DIM 3000
#define FT (F_DIM * T_DIM)            // 384000 floats per sample
#define CHUNK 128                     // floats per wave chunk (32 lanes * 4)
#define NCHUNK (FT / CHUNK)           // 3000 chunks per sample
#define RED_BLOCKS_PER_SAMPLE 16
#define RED_THREADS 256
#define WAVES_PER_BLOCK (RED_THREADS / 32)
#define APPLY_THREADS 256
#define VEC_PER_SAMPLE (FT / 4)       // 96000 float4 per sample
#define APPLY_BLOCKS_Y 125            // 96000 / (256*125) = 3 iters/thread

typedef __attribute__((ext_vector_type(4))) float v4f;

// Exact vector type from the builtin's signature (per hipcc diagnostic):
// __attribute__((__vector_size__(4 * sizeof(int)))) int
typedef int b128_t __attribute__((vector_size(16)));
typedef __attribute__((address_space(1))) b128_t* gbl_b128_ptr;
typedef __attribute__((address_space(3))) b128_t* lds_b128_ptr;

// Order-preserving float<->uint key so we can use integer atomicMin.
__device__ __forceinline__ unsigned f2key(float f) {
  unsigned u = __float_as_uint(f);
  return (u & 0x80000000u) ? ~u : (u | 0x80000000u);
}
__device__ __forceinline__ float key2f(unsigned k) {
  unsigned u = (k & 0x80000000u) ? (k ^ 0x80000000u) : ~k;
  return __uint_as_float(u);
}

// CDNA5 async global->LDS path (guarded so a missing builtin falls back cleanly)
#if defined(__gfx1250__) && \
    __has_builtin(__builtin_amdgcn_global_load_async_to_lds_b128) && \
    __has_builtin(__builtin_amdgcn_s_wait_asynccnt)
#define USE_ASYNC_LDS 1
#else
#define USE_ASYNC_LDS 0
#endif

__global__ void init_mins_kernel(unsigned* __restrict__ mins) {
  int i = threadIdx.x;
  if (i < B_DIM) mins[i] = 0xFFFFFFFFu;  // key(+inf)
}

__global__ __launch_bounds__(RED_THREADS)
void reduce_min_kernel(const float* __restrict__ x, unsigned* __restrict__ mins) {
  const int sample = blockIdx.x / RED_BLOCKS_PER_SAMPLE;
  const int blk    = blockIdx.x % RED_BLOCKS_PER_SAMPLE;
  const float* base = x + (size_t)sample * FT;

  const int lane  = threadIdx.x & 31;
  const int wave  = threadIdx.x >> 5;
  const int gwave = blk * WAVES_PER_BLOCK + wave;             // wave id within sample
  const int stride = RED_BLOCKS_PER_SAMPLE * WAVES_PER_BLOCK; // 128 waves / sample

  v4f m4 = {INFINITY, INFINITY, INFINITY, INFINITY};

#if USE_ASYNC_LDS
  __shared__ __align__(16) float lds[WAVES_PER_BLOCK][2][CHUNK];
  int c = gwave;
  int cur = 0;
  if (c < NCHUNK) {
    float* gp = const_cast<float*>(base + (size_t)c * CHUNK + lane * 4);
    __builtin_amdgcn_global_load_async_to_lds_b128(
        (gbl_b128_ptr)(b128_t*)gp,
        (lds_b128_ptr)(b128_t*)&lds[wave][0][lane * 4],
        0, 0);
  }
  for (; c < NCHUNK; c += stride) {
    const int cn = c + stride;
    if (cn < NCHUNK) {
      float* gp = const_cast<float*>(base + (size_t)cn * CHUNK + lane * 4);
      __builtin_amdgcn_global_load_async_to_lds_b128(
          (gbl_b128_ptr)(b128_t*)gp,
          (lds_b128_ptr)(b128_t*)&lds[wave][cur ^ 1][lane * 4],
          0, 0);
      __builtin_amdgcn_s_wait_asynccnt(1);   // previous chunk landed in LDS
    } else {
      __builtin_amdgcn_s_wait_asynccnt(0);
    }
    const v4f v = *(const v4f*)&lds[wave][cur][lane * 4];   // ds_load_b128
    m4.x = fminf(m4.x, v.x);
    m4.y = fminf(m4.y, v.y);
    m4.z = fminf(m4.z, v.z);
    m4.w = fminf(m4.w, v.w);
    cur ^= 1;
  }
#else
  for (int c = gwave; c < NCHUNK; c += stride) {
    const v4f v = *(const v4f*)(base + (size_t)c * CHUNK + lane * 4);
    m4.x = fminf(m4.x, v.x);
    m4.y = fminf(m4.y, v.y);
    m4.z = fminf(m4.z, v.z);
    m4.w = fminf(m4.w, v.w);
  }
#endif

  float m = fminf(fminf(m4.x, m4.y), fminf(m4.z, m4.w));
  #pragma unroll
  for (int off = 16; off > 0; off >>= 1)
    m = fminf(m, __shfl_xor(m, off, 32));
  if (lane == 0) atomicMin(&mins[sample], f2key(m));
}

__global__ __launch_bounds__(APPLY_THREADS)
void apply_mask_kernel(const float* __restrict__ x,
                       const int* __restrict__ f0, const int* __restrict__ fw,
                       const int* __restrict__ t0, const int* __restrict__ tw,
                       const unsigned* __restrict__ mins,
                       float* __restrict__ out) {
  const int b = blockIdx.x;
  // per-sample scalars (uniform per block -> SGPRs)
  const int f0a = f0[2 * b],       f0b = f0[2 * b + 1];
  const int f1a = f0a + fw[2 * b], f1b = f0b + fw[2 * b + 1];
  const int t0a = t0[2 * b],       t0b = t0[2 * b + 1];
  const int t1a = t0a + tw[2 * b], t1b = t0b + tw[2 * b + 1];
  const float fill = key2f(mins[b]);
  const size_t sbase = (size_t)b * FT;
  const int step = gridDim.y * blockDim.x;

  for (int v = blockIdx.y * blockDim.x + threadIdx.x; v < VEC_PER_SAMPLE; v += step) {
    const int e = v * 4;          // T_DIM % 4 == 0: a vec4 never crosses a row
    const int f = e / T_DIM;
    int t = e - f * T_DIM;

    if (v + step < VEC_PER_SAMPLE)
      __builtin_prefetch(x + sbase + (size_t)(v + step) * 4, 0, 0); // global_prefetch

    const bool fm = (f >= f0a && f < f1a) || (f >= f0b && f < f1b);
    const v4f val = __builtin_nontemporal_load((const v4f*)(x + sbase + e));
    v4f o;
    bool tm;
    tm = fm || (t >= t0a && t < t1a) || (t >= t0b && t < t1b); o.x = tm ? fill : val.x; ++t;
    tm = fm || (t >= t0a && t < t1a) || (t >= t0b && t < t1b); o.y = tm ? fill : val.y; ++t;
    tm = fm || (t >= t0a && t < t1a) || (t >= t0b && t < t1b); o.z = tm ? fill : val.z; ++t;
    tm = fm || (t >= t0a && t < t1a) || (t >= t0b && t < t1b); o.w = tm ? fill : val.w;
    __builtin_nontemporal_store(o, (v4f*)(out + sbase + e));
  }
}

extern "C" void kernel_launch(void* const* d_in, const int* in_sizes, int n_in,
                              void* d_out, int out_size, void* d_ws, size_t ws_size,
                              hipStream_t stream) {
  const float* x  = (const float*)d_in[0];
  const int*   f0 = (const int*)d_in[1];
  const int*   fw = (const int*)d_in[2];
  const int*   t0 = (const int*)d_in[3];
  const int*   tw = (const int*)d_in[4];
  float* out = (float*)d_out;
  unsigned* mins = (unsigned*)d_ws;   // 64 * 4 bytes of scratch

  init_mins_kernel<<<1, 64, 0, stream>>>(mins);
  reduce_min_kernel<<<B_DIM * RED_BLOCKS_PER_SAMPLE, RED_THREADS, 0, stream>>>(x, mins);
  dim3 grid(B_DIM, APPLY_BLOCKS_Y);
  apply_mask_kernel<<<grid, APPLY_THREADS, 0, stream>>>(x, f0, fw, t0, tw, mins, out);
}